// MVDRBeamformer_63625645523502
// MI455X (gfx1250) — compile-verified
//
#include <hip/hip_runtime.h>
#include <math.h>

// MVDR beamformer for MI455X (gfx1250, wave32).
//
// Covariance phase uses V_WMMA_F32_16X16X4_F32: A = [Y*speech_mask ; Y*noise_mask]
// (16 x K), B = [Y^H | Y^H] (K x 16). One 16x16 f32 accumulator pair (Re, Im)
// yields Phi_ss in quadrant [0:8,0:8] and Phi_nn in [8:16,8:16] at once.
// Complex product sum_t Yw * conj(Y):
//   Re += A_re.B_re + A_im.B_im ;  Im += A_im.B_re + (-A_re).B_im
//
// C/D layout (ISA 7.12.2): lane L<16 -> (M=vgpr, N=L); L>=16 -> (M=vgpr+8, N=L-16).
// A 16x4 layout: lane L holds row (L&15), VGPR j holds K = (L<16?0:2)+j  -> a
// contiguous float2 load per lane from the T-contiguous spectra.

typedef __attribute__((ext_vector_type(2))) float v2f;
typedef __attribute__((ext_vector_type(8))) float v8f;

#define NMIC   8
#define BATCH  8
#define FBINS  257
#define TLEN   2000
#define NW     4                       // waves per block
#define CPW    (TLEN / 4 / NW)         // 125 K=4 chunks per wave

__global__ __launch_bounds__(128)
void mvdr_wmma_kernel(const float* __restrict__ spec_re,
                      const float* __restrict__ spec_im,
                      const float* __restrict__ smask,
                      const float* __restrict__ nmask,
                      float* __restrict__ out)
{
    const int f    = blockIdx.x;
    const int b    = blockIdx.y;
    const int tid  = threadIdx.x;
    const int lane = tid & 31;
    const int wave = tid >> 5;

    __shared__ float ACCr[16][16];
    __shared__ float ACCi[16][16];
    __shared__ float msums[2];                 // [0]=speech, [1]=noise
    __shared__ float Ar[8][8], Ai[8][8];       // Phi_nn -> in-place inverse
    __shared__ float Sr[8][8], Si[8][8];       // Phi_ss
    __shared__ float Gr[8][8], Gi[8][8];       // Phi_nn^-1 * Phi_ss
    __shared__ float Wr[8],    Wi[8];          // MVDR weights

    // ---- zero reduction scratch -------------------------------------------
    for (int i = tid; i < 256; i += 128) { (&ACCr[0][0])[i] = 0.f; (&ACCi[0][0])[i] = 0.f; }
    if (tid < 2) msums[tid] = 0.f;
    __syncthreads();

    // ---- phase 1: covariance accumulation via WMMA ------------------------
    const int  row  = lane & 15;               // A row / B col index (0..15)
    const int  mic  = row & 7;                 // physical mic
    const int  koff = (lane < 16) ? 0 : 2;     // K sub-offset for this lane
    const long specOff = (((long)b * NMIC + mic) * FBINS + f) * TLEN;
    const float* pR = spec_re + specOff;
    const float* pI = spec_im + specOff;
    const float* pM = ((row < 8) ? smask : nmask) + ((long)b * FBINS + f) * TLEN;

    v8f accRe = {0.f,0.f,0.f,0.f,0.f,0.f,0.f,0.f};
    v8f accIm = {0.f,0.f,0.f,0.f,0.f,0.f,0.f,0.f};
    float msum = 0.f;

    const int c0 = wave * CPW;
    for (int c = c0; c < c0 + CPW; ++c) {
        const int t = c * 4 + koff;                                  // even -> 8B aligned
        const float2 r2 = *reinterpret_cast<const float2*>(pR + t);
        const float2 i2 = *reinterpret_cast<const float2*>(pI + t);
        const float2 mk = *reinterpret_cast<const float2*>(pM + t);
        msum += mk.x + mk.y;
        v2f aRe  = {  r2.x * mk.x,  r2.y * mk.y };
        v2f aIm  = {  i2.x * mk.x,  i2.y * mk.y };
        v2f aReN = { -aRe.x,       -aRe.y       };
        v2f bRe  = {  r2.x,         r2.y        };
        v2f bIm  = {  i2.x,         i2.y        };
        accRe = __builtin_amdgcn_wmma_f32_16x16x4_f32(false, aRe,  false, bRe, (short)0, accRe, false, false);
        accRe = __builtin_amdgcn_wmma_f32_16x16x4_f32(false, aIm,  false, bIm, (short)0, accRe, false, false);
        accIm = __builtin_amdgcn_wmma_f32_16x16x4_f32(false, aIm,  false, bRe, (short)0, accIm, false, false);
        accIm = __builtin_amdgcn_wmma_f32_16x16x4_f32(false, aReN, false, bIm, (short)0, accIm, false, false);
    }

    // ---- cross-wave reduction (ds_add_f32 atomics) ------------------------
    {
        const int mb = (lane < 16) ? 0 : 8;    // M base for this lane group
        const int n  = lane & 15;              // N column
#pragma unroll
        for (int v = 0; v < 8; ++v) {
            atomicAdd(&ACCr[mb + v][n], accRe[v]);
            atomicAdd(&ACCi[mb + v][n], accIm[v]);
        }
        float sp = __shfl(msum, 0) + __shfl(msum, 16);   // speech: lanes 0 & 16
        float np = __shfl(msum, 8) + __shfl(msum, 24);   // noise : lanes 8 & 24
        if (lane == 0) { atomicAdd(&msums[0], sp); atomicAdd(&msums[1], np); }
    }
    __syncthreads();

    // ---- phase 2: normalize + diagonal load -------------------------------
    if (tid < 64) {
        const int m = tid >> 3, n = tid & 7;
        const float is = 1.f / (msums[0] + 1e-8f);
        const float in_ = 1.f / (msums[1] + 1e-8f);
        Sr[m][n] = ACCr[m][n] * is;
        Si[m][n] = ACCi[m][n] * is;
        Ar[m][n] = ACCr[m + 8][n + 8] * in_ + ((m == n) ? 1e-5f : 0.f);
        Ai[m][n] = ACCi[m + 8][n + 8] * in_;
    }
    __syncthreads();

    // ---- in-place Gauss-Jordan inverse of Phi_nn (HPD, no pivoting) -------
    for (int k = 0; k < 8; ++k) {
        const int m = tid >> 3, n = tid & 7;
        float cr=0.f, ci=0.f, rr=0.f, ri=0.f, sr=0.f, si=0.f, ipr=0.f, ipi=0.f;
        if (tid < 64) {
            float pr = Ar[k][k], pi = Ai[k][k];
            float idet = 1.f / (pr * pr + pi * pi);
            ipr = pr * idet; ipi = -pi * idet;
            cr = Ar[m][k]; ci = Ai[m][k];
            rr = Ar[k][n]; ri = Ai[k][n];
            sr = Ar[m][n]; si = Ai[m][n];
        }
        __syncthreads();
        if (tid < 64) {
            float nrr = rr * ipr - ri * ipi;
            float nri = rr * ipi + ri * ipr;
            float orv, oiv;
            if (m == k && n == k)  { orv = ipr; oiv = ipi; }
            else if (m == k)       { orv = nrr; oiv = nri; }
            else if (n == k)       { orv = -(cr * ipr - ci * ipi); oiv = -(cr * ipi + ci * ipr); }
            else                   { orv = sr - (cr * nrr - ci * nri); oiv = si - (cr * nri + ci * nrr); }
            Ar[m][n] = orv; Ai[m][n] = oiv;
        }
        __syncthreads();
    }

    // ---- G = Phi_nn^-1 * Phi_ss -------------------------------------------
    if (tid < 64) {
        const int m = tid >> 3, n = tid & 7;
        float gr = 0.f, gi = 0.f;
#pragma unroll
        for (int j = 0; j < 8; ++j) {
            float a = Ar[m][j], bb = Ai[m][j], c = Sr[j][n], d = Si[j][n];
            gr += a * c - bb * d;
            gi += a * d + bb * c;
        }
        Gr[m][n] = gr; Gi[m][n] = gi;
    }
    __syncthreads();

    // ---- wave 0: power iteration + MVDR weights (register/shuffle only) ---
    if (wave == 0) {
        const int r = lane & 7;                 // row this lane owns (lanes>=8 shadow)
        float vre = 1.f, vim = 0.f;
        for (int it = 0; it < 30; ++it) {
            float nr = 0.f, ni = 0.f;
#pragma unroll
            for (int n = 0; n < 8; ++n) {
                float vnr = __shfl(vre, n), vni = __shfl(vim, n);
                float gr = Gr[r][n], gi = Gi[r][n];
                nr += gr * vnr - gi * vni;
                ni += gr * vni + gi * vnr;
            }
            float s = (lane < 8) ? (nr * nr + ni * ni) : 0.f;
            s += __shfl_xor(s, 1); s += __shfl_xor(s, 2); s += __shfl_xor(s, 4);
            float inv = 1.f / (sqrtf(s) + 1e-12f);
            vre = nr * inv; vim = ni * inv;
        }
        // rtf = v / (v[0] + 1e-8)
        float v0r = __shfl(vre, 0) + 1e-8f;
        float v0i = __shfl(vim, 0);
        float id0 = 1.f / (v0r * v0r + v0i * v0i);
        float rtr = (vre * v0r + vim * v0i) * id0;
        float rti = (vim * v0r - vre * v0i) * id0;
        // d = Phi_nn^-1 * rtf
        float dr = 0.f, di = 0.f;
#pragma unroll
        for (int n = 0; n < 8; ++n) {
            float rnr = __shfl(rtr, n), rni = __shfl(rti, n);
            float a = Ar[r][n], bb = Ai[r][n];
            dr += a * rnr - bb * rni;
            di += a * rni + bb * rnr;
        }
        // denom = Re(rtf^H d) + 1e-8
        float p = (lane < 8) ? (rtr * dr + rti * di) : 0.f;
        p += __shfl_xor(p, 1); p += __shfl_xor(p, 2); p += __shfl_xor(p, 4);
        float idn = 1.f / (p + 1e-8f);
        if (lane < 8) { Wr[lane] = dr * idn; Wi[lane] = di * idn; }
    }
    __syncthreads();

    // ---- phase 3: enhanced = w^H Y (coalesced over t) ---------------------
    const long yb = (((long)b * NMIC) * FBINS + f) * TLEN;
    float* po = out + 2 * (((long)b * FBINS + f) * TLEN);
    for (int t = tid; t < TLEN; t += 128) {
        float er = 0.f, ei = 0.f;
#pragma unroll
        for (int m = 0; m < NMIC; ++m) {
            float yr = spec_re[yb + (long)m * FBINS * TLEN + t];
            float yi = spec_im[yb + (long)m * FBINS * TLEN + t];
            float a = Wr[m], bb = Wi[m];
            // conj(w) * y
            er += a * yr + bb * yi;
            ei += a * yi - bb * yr;
        }
        po[2 * t]     = er;
        po[2 * t + 1] = ei;
    }
}

extern "C" void kernel_launch(void* const* d_in, const int* in_sizes, int n_in,
                              void* d_out, int out_size, void* d_ws, size_t ws_size,
                              hipStream_t stream) {
    const float* spec_re = (const float*)d_in[0];
    const float* spec_im = (const float*)d_in[1];
    const float* smask   = (const float*)d_in[2];
    const float* nmask   = (const float*)d_in[3];
    float* out = (float*)d_out;   // complex64 (B,F,T) flattened as interleaved re/im

    dim3 grid(FBINS, BATCH);      // one block per (b, f)
    mvdr_wmma_kernel<<<grid, 128, 0, stream>>>(spec_re, spec_im, smask, nmask, out);
}